// DomainAdaptationModel_87746181857787
// MI455X (gfx1250) — compile-verified
//
#include <hip/hip_runtime.h>
#include <hip/hip_bf16.h>

#define N_SRC 20000
#define N_TGT 20000
#define NN    40000
#define EDGES 320000
#define DD    512
#define HH    256

typedef __attribute__((ext_vector_type(2))) float v2f;
typedef __attribute__((ext_vector_type(8))) float v8f;

// ---------------------------------------------------------------------------
// Kernel 1: h_neigh[dst] += features[src]   (segment_sum via fp32 HW atomics)
// 128 threads per edge, float4 per thread. h_neigh (82MB) is L2-resident.
// ---------------------------------------------------------------------------
__global__ void segsum_kernel(const float* __restrict__ fs,
                              const float* __restrict__ ft,
                              const int* __restrict__ src,
                              const int* __restrict__ dst,
                              float* __restrict__ hneigh) {
  int tid = blockIdx.x * blockDim.x + threadIdx.x;
  int e = tid >> 7;            // edge index (128 threads/edge)
  int p = (tid & 127) << 2;    // float offset within row
  if (e >= EDGES) return;
  int s = src[e];
  int d = dst[e];
  const float* row = (s < N_SRC) ? (fs + (size_t)s * DD)
                                 : (ft + (size_t)(s - N_SRC) * DD);
  float4 v = *(const float4*)(row + p);
  float* out = hneigh + (size_t)d * DD + p;
  unsafeAtomicAdd(out + 0, v.x);
  unsafeAtomicAdd(out + 1, v.y);
  unsafeAtomicAdd(out + 2, v.z);
  unsafeAtomicAdd(out + 3, v.w);
}

// ---------------------------------------------------------------------------
// Kernel 2: h = theta1*relu(X@W+b) + theta2*relu(Hn@W+b)
// One wave32 per 16x16 output tile; fp32 WMMA 16x16x4, K = 512.
// A layout (32-bit 16x4): lane m (both halves) holds row m; half selects K
// pair -> contiguous float2 load at (row, k0 + 2*half).
// B layout: vgpr0 = K rows {0,2}, vgpr1 = K rows {1,3}, N striped on lanes.
// ---------------------------------------------------------------------------
__global__ void gnn_kernel(const float* __restrict__ fs,
                           const float* __restrict__ ft,
                           const float* __restrict__ hneigh,
                           const float* __restrict__ W,
                           const float* __restrict__ bias,
                           const float* __restrict__ th1p,
                           const float* __restrict__ th2p,
                           float* __restrict__ h) {
  const int lane = threadIdx.x & 31;
  const int wave = threadIdx.x >> 5;
  const int tile = blockIdx.x * 8 + wave;      // 80000 tiles exactly
  const int rt = tile >> 5;                    // 2500 row tiles
  const int ct = tile & 31;                    // 32 col tiles
  const int row0 = rt * 16;
  const int n0 = ct * 16;
  const int half = lane >> 4;
  const int m = lane & 15;

  const float th1 = th1p[0];
  const float th2 = th2p[0];

  const int arow = row0 + m;                   // < 40000, N_SRC%16==0
  const float* xrow = (arow < N_SRC) ? (fs + (size_t)arow * DD)
                                     : (ft + (size_t)(arow - N_SRC) * DD);
  const float* nrow = hneigh + (size_t)arow * DD;

  v8f accx = {};
  v8f accn = {};
#pragma unroll 4
  for (int k0 = 0; k0 < DD; k0 += 4) {
    const int ka = k0 + 2 * half;
    v2f ax = *(const v2f*)(xrow + ka);
    v2f an = *(const v2f*)(nrow + ka);
    v2f bb;
    bb.x = W[(size_t)ka * DD + n0 + m];
    bb.y = W[(size_t)(ka + 1) * DD + n0 + m];
    accx = __builtin_amdgcn_wmma_f32_16x16x4_f32(false, ax, false, bb,
                                                 (short)0, accx, false, false);
    accn = __builtin_amdgcn_wmma_f32_16x16x4_f32(false, an, false, bb,
                                                 (short)0, accn, false, false);
  }

  const float bn = bias[n0 + m];
#pragma unroll
  for (int j = 0; j < 8; ++j) {
    const int r = row0 + j + 8 * half;         // C layout: vgpr j -> row j/j+8
    float vx = accx[j] + bn; vx = vx > 0.f ? vx : 0.f;
    float vn = accn[j] + bn; vn = vn > 0.f ? vn : 0.f;
    h[(size_t)r * DD + n0 + m] = th1 * vx + th2 * vn;
  }
}

// ---------------------------------------------------------------------------
// Kernel 3: y = relu(h@W1+b1)@W2 + b2
// One wave32 per 16-row group; 16 hidden col-tiles via WMMA into LDS, then
// tiny 256->2 projection per lane. 4 waves/block, 64KB LDS.
// ---------------------------------------------------------------------------
__global__ void cls_kernel(const float* __restrict__ h,
                           const float* __restrict__ W1,
                           const float* __restrict__ b1,
                           const float* __restrict__ W2,
                           const float* __restrict__ b2,
                           float* __restrict__ out) {
  __shared__ float lds[4][16][HH];             // 64 KB
  const int lane = threadIdx.x & 31;
  const int wave = threadIdx.x >> 5;
  const int tile = blockIdx.x * 4 + wave;      // 2500 tiles exactly
  const int row0 = tile * 16;
  const int half = lane >> 4;
  const int m = lane & 15;

  const float* arow = h + (size_t)(row0 + m) * DD;

  for (int ct = 0; ct < HH / 16; ++ct) {
    const int n0 = ct * 16;
    v8f acc = {};
#pragma unroll 4
    for (int k0 = 0; k0 < DD; k0 += 4) {
      const int ka = k0 + 2 * half;
      v2f a = *(const v2f*)(arow + ka);
      v2f bb;
      bb.x = W1[(size_t)ka * HH + n0 + m];
      bb.y = W1[(size_t)(ka + 1) * HH + n0 + m];
      acc = __builtin_amdgcn_wmma_f32_16x16x4_f32(false, a, false, bb,
                                                  (short)0, acc, false, false);
    }
    const float bn = b1[n0 + m];
#pragma unroll
    for (int j = 0; j < 8; ++j) {
      float v = acc[j] + bn;
      lds[wave][j + 8 * half][n0 + m] = v > 0.f ? v : 0.f;
    }
  }
  __syncthreads();

  // 32 lanes: 16 rows x 2 classes
  const int mm = lane >> 1;
  const int c = lane & 1;
  float y = b2[c];
#pragma unroll 4
  for (int n = 0; n < HH; ++n)
    y += lds[wave][mm][n] * W2[n * 2 + c];
  out[(size_t)(row0 + mm) * 2 + c] = y;
}

// ---------------------------------------------------------------------------
extern "C" void kernel_launch(void* const* d_in, const int* in_sizes, int n_in,
                              void* d_out, int out_size, void* d_ws, size_t ws_size,
                              hipStream_t stream) {
  const float* fs  = (const float*)d_in[0];
  const float* ft  = (const float*)d_in[1];
  const int*   src = (const int*)d_in[2];
  const int*   dst = (const int*)d_in[3];
  const float* Wg  = (const float*)d_in[4];
  const float* bg  = (const float*)d_in[5];
  const float* th1 = (const float*)d_in[6];
  const float* th2 = (const float*)d_in[7];
  const float* W1  = (const float*)d_in[8];
  const float* b1  = (const float*)d_in[9];
  const float* W2  = (const float*)d_in[10];
  const float* b2  = (const float*)d_in[11];
  float* out = (float*)d_out;

  float* hneigh = (float*)d_ws;                      // 40000*512*4 = 82MB
  float* h      = hneigh + (size_t)NN * DD;          // 82MB more

  // zero the segment-sum accumulator (graph-capturable)
  hipMemsetAsync(hneigh, 0, (size_t)NN * DD * sizeof(float), stream);

  // 1) segment sum: 320000 edges * 128 threads = 40.96M threads
  segsum_kernel<<<(EDGES * 128) / 256, 256, 0, stream>>>(fs, ft, src, dst, hneigh);

  // 2) fused GNN layer: 2500*32 = 80000 tiles, 8 waves/block
  gnn_kernel<<<80000 / 8, 256, 0, stream>>>(fs, ft, hneigh, Wg, bg, th1, th2, h);

  // 3) fused classifier: 2500 tiles, 4 waves/block
  cls_kernel<<<2500 / 4, 128, 0, stream>>>(h, W1, b1, W2, b2, out);
}